// HolonomyCrusher_12661563589253
// MI455X (gfx1250) — compile-verified
//
#include <hip/hip_runtime.h>
#include <hip/hip_bf16.h>
#include <math.h>

// ---------------- CDNA5 WMMA fp32 helper ----------------
typedef __attribute__((ext_vector_type(2))) float v2f;
typedef __attribute__((ext_vector_type(8))) float v8f;

__device__ __forceinline__ v8f wmma_f32_16x16x4(v2f a, v2f b, v8f c) {
    // 8 args: (neg_a, A, neg_b, B, c_mod, C, reuse_a, reuse_b)
    return __builtin_amdgcn_wmma_f32_16x16x4_f32(false, a, false, b, (short)0, c,
                                                 false, false);
}

// ---------------- constants ----------------
#define BB 16
#define VV 32000
#define DD 1024
#define FF 128
#define RR 16
#define TK 64
#define WW 8
#define NSTEPS 4
#define STR 132   // padded LDS row/col stride: 132 mod 64 banks = 4 -> conflict-free

// ---------------- top-k (block per batch row) ----------------
__global__ void topk_kernel(const float* __restrict__ logits,
                            int* __restrict__ idx_out) {
    extern __shared__ float smem[];
    float* vals = smem;                 // VV floats
    float* rv = smem + VV;              // 512
    int* ri = (int*)(rv + 512);         // 512
    int b = blockIdx.x, tid = threadIdx.x;
    const float* row = logits + (size_t)b * VV;
    for (int i = tid; i < VV; i += 512) vals[i] = row[i];
    __syncthreads();
    for (int it = 0; it < TK; ++it) {
        float best = -INFINITY; int bi = 0;
        for (int i = tid; i < VV; i += 512) {
            float v = vals[i];
            if (v > best) { best = v; bi = i; }
        }
        rv[tid] = best; ri[tid] = bi;
        __syncthreads();
        for (int o = 256; o > 0; o >>= 1) {
            if (tid < o) {
                if (rv[tid + o] > rv[tid] ||
                    (rv[tid + o] == rv[tid] && ri[tid + o] < ri[tid])) {
                    rv[tid] = rv[tid + o]; ri[tid] = ri[tid + o];
                }
            }
            __syncthreads();
        }
        if (tid == 0) { idx_out[b * TK + it] = ri[0]; vals[ri[0]] = -INFINITY; }
        __syncthreads();
    }
}

// ---------------- specialized fp32 WMMA GEMM ----------------
// Y[M x N] = act( XM @ W + bias )
// MODE: 0 plain, 1 mix ((1-t)Xa + t*Xb), 2 gather (Xa[rowidx[m]])
// ACT:  0 none, 1 gelu(erf), 2 tanh*scale
// NT:   16-wide N tiles per wave (one A float2 load feeds NT WMMAs)
template <int MODE, int ACT, int NT>
__global__ void gemm_kernel(const float* __restrict__ Xa,
                            const float* __restrict__ Xb, float t,
                            const int* __restrict__ rowidx,
                            const float* __restrict__ W,
                            const float* __restrict__ bias,
                            float* __restrict__ Y, int K, int N,
                            const float* __restrict__ scale_ptr) {
    const int lane = threadIdx.x;            // 0..31 (one wave per block)
    const int tm = blockIdx.x, tn = blockIdx.y;
    const int mrow = tm * 16 + (lane & 15);
    int rA = (MODE == 2) ? rowidx[mrow] : mrow;
    const int koff = (lane >> 4) * 2;
    const int ncol0 = tn * (16 * NT) + (lane & 15);

    const float* aptr = Xa + (size_t)rA * K + koff;
    const float* bptr = W + (size_t)koff * N + ncol0;
    const float* xptr = (MODE == 1) ? (Xb + (size_t)mrow * K + koff) : nullptr;
    const float omt = 1.0f - t;

    v8f acc[NT];
#pragma unroll
    for (int nt = 0; nt < NT; ++nt) acc[nt] = v8f{};

    for (int k = 0; k < K; k += 4) {
        v2f a;
        if (MODE == 1) {
            v2f va = *(const v2f*)aptr;
            v2f vb = *(const v2f*)xptr;
            a.x = omt * va.x + t * vb.x;
            a.y = omt * va.y + t * vb.y;
            xptr += 4;
        } else {
            a = *(const v2f*)aptr;
        }
        aptr += 4;
        __builtin_prefetch((const void*)(bptr + (size_t)32 * N), 0, 0);
#pragma unroll
        for (int nt = 0; nt < NT; ++nt) {
            v2f b;
            b.x = bptr[nt * 16];
            b.y = bptr[(size_t)N + nt * 16];
            acc[nt] = wmma_f32_16x16x4(a, b, acc[nt]);
        }
        bptr += (size_t)4 * N;
    }

    const float sc = (ACT == 2) ? *scale_ptr : 1.0f;
#pragma unroll
    for (int nt = 0; nt < NT; ++nt) {
        const int ncol = ncol0 + nt * 16;
        const float bv = bias[ncol];
#pragma unroll
        for (int j = 0; j < 8; ++j) {
            int m = tm * 16 + j + 8 * (lane >> 4);
            float v = acc[nt][j] + bv;
            if (ACT == 1) v = 0.5f * v * (1.0f + erff(v * 0.70710678118654752f));
            else if (ACT == 2) v = tanhf(v) * sc;
            Y[(size_t)m * N + ncol] = v;
        }
    }
}

// ---------------- layernorm (block per row of D=1024) ----------------
// aux==null -> Y=ln(x); aux!=null -> Y = aux[row/64] + mix*ln(x)
__global__ void ln_kernel(const float* __restrict__ X,
                          const float* __restrict__ g,
                          const float* __restrict__ b, float* __restrict__ Y,
                          const float* __restrict__ aux, float mix) {
    __shared__ float red[256];
    int row = blockIdx.x, tid = threadIdx.x;
    const float* x = X + (size_t)row * DD;
    float s = 0.f;
    for (int i = tid; i < DD; i += 256) s += x[i];
    red[tid] = s; __syncthreads();
    for (int o = 128; o > 0; o >>= 1) { if (tid < o) red[tid] += red[tid + o]; __syncthreads(); }
    float mean = red[0] / DD;
    __syncthreads();
    float s2 = 0.f;
    for (int i = tid; i < DD; i += 256) { float d = x[i] - mean; s2 += d * d; }
    red[tid] = s2; __syncthreads();
    for (int o = 128; o > 0; o >>= 1) { if (tid < o) red[tid] += red[tid + o]; __syncthreads(); }
    float rstd = rsqrtf(red[0] / DD + 1e-5f);
    const float* auxrow = aux ? aux + (size_t)(row / TK) * DD : nullptr;
    for (int i = tid; i < DD; i += 256) {
        float y = (x[i] - mean) * rstd * g[i] + b[i];
        Y[(size_t)row * DD + i] = auxrow ? auxrow[i] + mix * y : y;
    }
}

// ---------------- transport step update (block per element) ----------------
// A = sum_r c_r G_r ; dT = I + A*dt + A@A*(0.5 dt^2) ; T <- dT @ T
// LDS layouts (stride 132, bank-conflict-free for both operand shapes):
//   sA  : A row-major     (A-operand of A@A, elementwise reads)
//   sAc : A column-major  (B-operand of A@A -> contiguous (kk,kk+1) b64 load)
//   sDT : dT row-major    (A-operand of dT@T)
//   sTc : T column-major  (B-operand of dT@T)
// 4 * 128*132 floats = 264KB <= 320KB WGP LDS.
// 256 threads = 8 waves; wave w owns row-block w: one A float2 feeds 8 WMMAs.
__global__ void transport_update_kernel(const float* __restrict__ Cc,
                                        const float* __restrict__ gen,
                                        float* __restrict__ Tg, float dt) {
    extern __shared__ float sm[];
    float* sA  = sm;                  // FF*STR
    float* sAc = sm + FF * STR;       // FF*STR
    float* sDT = sm + 2 * FF * STR;   // FF*STR
    float* sTc = sm + 3 * FF * STR;   // FF*STR
    const int m = blockIdx.x, tid = threadIdx.x;
    float c[RR];
#pragma unroll
    for (int r = 0; r < RR; ++r) c[r] = Cc[m * RR + r];
    {   // A = sum_r c_r * G_r ; write row-major + column-major copies
        const float4* g4 = (const float4*)gen;
        for (int i = tid; i < FF * FF / 4; i += 256) {
            int row = i >> 5, c4 = (i & 31) * 4;
            float4 a = make_float4(0.f, 0.f, 0.f, 0.f);
#pragma unroll
            for (int r = 0; r < RR; ++r) {
                float4 gv = g4[(size_t)r * (FF * FF / 4) + i];
                a.x += c[r] * gv.x; a.y += c[r] * gv.y;
                a.z += c[r] * gv.z; a.w += c[r] * gv.w;
            }
            *(float4*)&sA[row * STR + c4] = a;
            sAc[(c4 + 0) * STR + row] = a.x;
            sAc[(c4 + 1) * STR + row] = a.y;
            sAc[(c4 + 2) * STR + row] = a.z;
            sAc[(c4 + 3) * STR + row] = a.w;
        }
    }
    {   // stage current T column-major
        const float4* Tg4 = (const float4*)(Tg + (size_t)m * FF * FF);
        for (int i = tid; i < FF * FF / 4; i += 256) {
            int row = i >> 5, c4 = (i & 31) * 4;
            float4 tv = Tg4[i];
            sTc[(c4 + 0) * STR + row] = tv.x;
            sTc[(c4 + 1) * STR + row] = tv.y;
            sTc[(c4 + 2) * STR + row] = tv.z;
            sTc[(c4 + 3) * STR + row] = tv.w;
        }
    }
    __syncthreads();
    const int wave = tid >> 5, lane = tid & 31;
    const int koff = (lane >> 4) * 2;
    const int mr = wave * 16 + (lane & 15);
    const float hdt2 = 0.5f * dt * dt;

    // ---- dT = I + A dt + (A@A) hdt2 : wave computes row-block `wave`
    {
        v8f acc[8];
#pragma unroll
        for (int tj = 0; tj < 8; ++tj) acc[tj] = v8f{};
        for (int k = 0; k < FF; k += 4) {
            int kk = k + koff;
            v2f a = *(const v2f*)&sA[mr * STR + kk];
#pragma unroll
            for (int tj = 0; tj < 8; ++tj) {
                int nc = tj * 16 + (lane & 15);
                v2f b = *(const v2f*)&sAc[nc * STR + kk];
                acc[tj] = wmma_f32_16x16x4(a, b, acc[tj]);
            }
        }
#pragma unroll
        for (int tj = 0; tj < 8; ++tj) {
            int nc = tj * 16 + (lane & 15);
#pragma unroll
            for (int j = 0; j < 8; ++j) {
                int rr = wave * 16 + j + 8 * (lane >> 4);
                float v = ((rr == nc) ? 1.f : 0.f) + sA[rr * STR + nc] * dt +
                          acc[tj][j] * hdt2;
                sDT[rr * STR + nc] = v;
            }
        }
    }
    __syncthreads();
    // ---- T <- dT @ T
    {
        float* Tout = Tg + (size_t)m * FF * FF;
        v8f acc[8];
#pragma unroll
        for (int tj = 0; tj < 8; ++tj) acc[tj] = v8f{};
        for (int k = 0; k < FF; k += 4) {
            int kk = k + koff;
            v2f a = *(const v2f*)&sDT[mr * STR + kk];
#pragma unroll
            for (int tj = 0; tj < 8; ++tj) {
                int nc = tj * 16 + (lane & 15);
                v2f b = *(const v2f*)&sTc[nc * STR + kk];
                acc[tj] = wmma_f32_16x16x4(a, b, acc[tj]);
            }
        }
#pragma unroll
        for (int tj = 0; tj < 8; ++tj) {
            int nc = tj * 16 + (lane & 15);
#pragma unroll
            for (int j = 0; j < 8; ++j) {
                int rr = wave * 16 + j + 8 * (lane >> 4);
                Tout[rr * FF + nc] = acc[tj][j];
            }
        }
    }
}

// ---------------- small utility kernels ----------------
__global__ void set_identity_kernel(float* __restrict__ T) {
    int m = blockIdx.x, tid = threadIdx.x;
    float* t = T + (size_t)m * FF * FF;
    for (int e = tid; e < FF * FF; e += 256)
        t[e] = ((e >> 7) == (e & 127)) ? 1.f : 0.f;
}
__global__ void copy_mats_kernel(float* __restrict__ dst,
                                 const float* __restrict__ src, int rep) {
    int m = blockIdx.x, tid = threadIdx.x;
    const float4* s = (const float4*)(src + (size_t)(m / rep) * FF * FF);
    float4* d = (float4*)(dst + (size_t)m * FF * FF);
    for (int i = tid; i < FF * FF / 4; i += 256) d[i] = s[i];
}
__global__ void bcast_rows_kernel(float* __restrict__ dst,
                                  const float* __restrict__ src, int rep) {
    int m = blockIdx.x, tid = threadIdx.x;
    const float* s = src + (size_t)(m / rep) * DD;
    float* d = dst + (size_t)m * DD;
    for (int i = tid; i < DD; i += 256) d[i] = s[i];
}
__global__ void dist_kernel(const float* __restrict__ T, float* __restrict__ dist) {
    __shared__ float red[256];
    int m = blockIdx.x, tid = threadIdx.x;
    const float* t = T + (size_t)m * FF * FF;
    float s = 0.f;
    for (int e = tid; e < FF * FF; e += 256) {
        float v = t[e] - (((e >> 7) == (e & 127)) ? 1.f : 0.f);
        s += v * v;
    }
    red[tid] = s; __syncthreads();
    for (int o = 128; o > 0; o >>= 1) { if (tid < o) red[tid] += red[tid + o]; __syncthreads(); }
    if (tid == 0) dist[m] = sqrtf(red[0]);
}
__global__ void out_base_kernel(const float* __restrict__ logits,
                                float* __restrict__ out, int n) {
    int i = blockIdx.x * 256 + threadIdx.x;
    if (i < n) out[i] = logits[i] + logf(1.0f + 1e-10f);
}
__global__ void crush_scatter_kernel(const float* __restrict__ logits,
                                     const int* __restrict__ idx,
                                     const float* __restrict__ nd,
                                     const float* __restrict__ cd,
                                     float* __restrict__ out) {
    int m = blockIdx.x * TK + threadIdx.x;   // 16 blocks x 64 threads
    int b = m / TK;
    float delta = nd[m] - cd[b];
    float z = -100.0f * (delta - 0.05f);
    float crush = 1.0f / (1.0f + expf(-z));
    int col = idx[m];
    out[(size_t)b * VV + col] = logits[(size_t)b * VV + col] + logf(crush + 1e-10f);
}

// ---------------- host orchestration ----------------
extern "C" void kernel_launch(void* const* d_in, const int* in_sizes, int n_in,
                              void* d_out, int out_size, void* d_ws, size_t ws_size,
                              hipStream_t stream) {
    const float* logits = (const float*)d_in[0];
    const float* hidden = (const float*)d_in[1];
    const float* path   = (const float*)d_in[2];   // (8,16,1024)
    const float* emb    = (const float*)d_in[3];
    const float* enc_W  = (const float*)d_in[4];
    const float* enc_b  = (const float*)d_in[5];
    const float* ln_g   = (const float*)d_in[6];
    const float* ln_b   = (const float*)d_in[7];
    const float* W1     = (const float*)d_in[8];
    const float* b1     = (const float*)d_in[9];
    const float* W2     = (const float*)d_in[10];
    const float* b2     = (const float*)d_in[11];
    const float* gen    = (const float*)d_in[12];
    const float* scale  = (const float*)d_in[13];
    float* out = (float*)d_out;

    const int MB = BB * TK;  // 1024 candidate rows
    // carve workspace
    char* p = (char*)d_ws;
    auto carve = [&](size_t bytes) { char* r = p; p += (bytes + 255) & ~size_t(255); return (void*)r; };
    int*   idx   = (int*)  carve((size_t)MB * 4);
    float* tmpA  = (float*)carve((size_t)MB * DD * 4);  // pre-LN scratch, reused as H
    float* curst = (float*)carve((size_t)BB * DD * 4);
    float* cand  = (float*)carve((size_t)MB * DD * 4);
    float* Cbuf  = (float*)carve((size_t)MB * RR * 4);
    float* Tsm   = (float*)carve((size_t)BB * FF * FF * 4);   // becomes P
    float* Tcur  = (float*)carve((size_t)BB * FF * FF * 4);
    float* Tbig  = (float*)carve((size_t)MB * FF * FF * 4);
    float* XSb   = (float*)carve((size_t)MB * DD * 4);
    float* XEb   = (float*)carve((size_t)MB * DD * 4);
    float* curd  = (float*)carve((size_t)BB * 4);
    float* newd  = (float*)carve((size_t)MB * 4);
    float* H = tmpA;

    const size_t ldsUpd = (size_t)4 * FF * STR * sizeof(float);   // 264KB
    const float dt = 1.0f / NSTEPS;

    auto transport = [&](int M, const float* xs, const float* xe, float* T) {
        for (int i = 0; i < NSTEPS; ++i) {
            float t = (i + 0.5f) / NSTEPS;
            // H = gelu(((1-t)xs + t*xe) @ W1 + b1)   [MODE=1 mix, ACT=1 gelu, NT=8]
            gemm_kernel<1, 1, 8><<<dim3(M / 16, DD / 128), 32, 0, stream>>>(
                xs, xe, t, nullptr, W1, b1, H, DD, DD, nullptr);
            // C = tanh(H @ W2 + b2) * scale          [MODE=0, ACT=2, NT=1]
            gemm_kernel<0, 2, 1><<<dim3(M / 16, 1), 32, 0, stream>>>(
                H, nullptr, 0.f, nullptr, W2, b2, Cbuf, DD, RR, scale);
            transport_update_kernel<<<M, 256, ldsUpd, stream>>>(Cbuf, gen, T, dt);
        }
    };

    // 1) top-k
    topk_kernel<<<BB, 512, (size_t)(VV + 1024) * 4, stream>>>(logits, idx);
    // 2) current_state = encode(hidden)
    gemm_kernel<0, 0, 8><<<dim3(BB / 16, DD / 128), 32, 0, stream>>>(
        hidden, nullptr, 0.f, nullptr, enc_W, enc_b, tmpA, DD, DD, nullptr);
    ln_kernel<<<BB, 256, 0, stream>>>(tmpA, ln_g, ln_b, curst, nullptr, 0.f);
    // 3) cand = curstate + 0.1 * encode(emb[top_idx])   [MODE=2 gather]
    gemm_kernel<2, 0, 8><<<dim3(MB / 16, DD / 128), 32, 0, stream>>>(
        emb, nullptr, 0.f, idx, enc_W, enc_b, tmpA, DD, DD, nullptr);
    ln_kernel<<<MB, 256, 0, stream>>>(tmpA, ln_g, ln_b, cand, curst, 0.1f);
    // 4) P = product of window-edge transports (T starts at I, each run left-multiplies)
    set_identity_kernel<<<BB, 256, 0, stream>>>(Tsm);
    for (int e = 0; e < WW - 1; ++e)
        transport(BB, path + (size_t)e * BB * DD, path + (size_t)(e + 1) * BB * DD, Tsm);
    // 5) cur = transport(path[7]->path[0]) @ P  (init T = P)
    copy_mats_kernel<<<BB, 256, 0, stream>>>(Tcur, Tsm, 1);
    transport(BB, path + (size_t)(WW - 1) * BB * DD, path, Tcur);
    dist_kernel<<<BB, 256, 0, stream>>>(Tcur, curd);
    // 6) Q = T_mid @ P  (init T = P[b], xs = path[7][b] bcast, xe = cand)
    copy_mats_kernel<<<MB, 256, 0, stream>>>(Tbig, Tsm, TK);
    bcast_rows_kernel<<<MB, 256, 0, stream>>>(XSb, path + (size_t)(WW - 1) * BB * DD, TK);
    transport(MB, XSb, cand, Tbig);
    // 7) new = T_last @ Q  (xs = cand, xe = path[0][b] bcast, T continues)
    bcast_rows_kernel<<<MB, 256, 0, stream>>>(XEb, path, TK);
    transport(MB, cand, XEb, Tbig);
    dist_kernel<<<MB, 256, 0, stream>>>(Tbig, newd);
    // 8) output
    out_base_kernel<<<(BB * VV + 255) / 256, 256, 0, stream>>>(logits, out, BB * VV);
    crush_scatter_kernel<<<BB, TK, 0, stream>>>(logits, idx, newd, curd, out);
    (void)in_sizes; (void)n_in; (void)out_size; (void)ws_size;
}